// TemplateEmbedding_65377992179721
// MI455X (gfx1250) — compile-verified
//
#include <hip/hip_runtime.h>
#include <math.h>

// MI455X / gfx1250 implementation.
// Stage A: conv tower entirely in LDS (192KB buffers, fits 320KB/WGP), emits
//          hoisted left/right projections li/rj into workspace (1 MB needed).
// Stage B: pair projection via V_WMMA_F32_16X16X4_F32, LDS-staged B operand,
//          LDS-transposed coalesced b128 output stores.
//          t2d reads and output writes are single-use streams -> non-temporal
//          hints so they don't sweep the 192MB L2 (keeps W2d/li/rj hot).

typedef __attribute__((ext_vector_type(2))) float v2f;
typedef __attribute__((ext_vector_type(4))) float v4f;
typedef __attribute__((ext_vector_type(8))) float v8f;

#define Lx    512
#define F1    46
#define F2    44
#define Cc    32
#define Kk    64
#define PWROW 109   // F2 + 2*C + 1

__device__ __forceinline__ float elu1(float x) {
    return x > 0.0f ? x : (__expf(x) - 1.0f);
}

// conv1d(C->C, k=3, pad=1) over LDS buffers, 256 threads
__device__ void conv3(const float* __restrict__ src, float* __restrict__ dst,
                      const float* __restrict__ W, int t)
{
    for (int idx = t; idx < Cc * Lx; idx += 256) {
        int c = idx >> 9, l = idx & (Lx - 1);
        const float* w = W + c * (Cc * 3);
        float acc = 0.f;
#pragma unroll 4
        for (int ci = 0; ci < Cc; ++ci) {
            const float* xr = src + (ci << 9);
            float x0 = (l > 0)      ? xr[l - 1] : 0.f;
            float x1 = xr[l];
            float x2 = (l < Lx - 1) ? xr[l + 1] : 0.f;
            acc = fmaf(w[ci * 3 + 0], x0, acc);
            acc = fmaf(w[ci * 3 + 1], x1, acc);
            acc = fmaf(w[ci * 3 + 2], x2, acc);
        }
        dst[idx] = acc;
    }
}

// per-channel mean / rstd over L=512 -> MV[0..31]=mean, MV[32..63]=rstd
__device__ void inorm_stats(const float* __restrict__ buf, float* RED,
                            float* RED2, float* MV, int t)
{
    int c = t >> 3, s = t & 7;
    const float* row = buf + (c << 9) + (s << 6);
    float sm = 0.f, sq = 0.f;
    for (int k = 0; k < 64; ++k) { float v = row[k]; sm += v; sq = fmaf(v, v, sq); }
    RED[t] = sm; RED2[t] = sq;
    __syncthreads();
    if (t < Cc) {
        float S = 0.f, Q = 0.f;
        for (int r = 0; r < 8; ++r) { S += RED[t * 8 + r]; Q += RED2[t * 8 + r]; }
        float m   = S * (1.0f / Lx);
        float var = Q * (1.0f / Lx) - m * m;
        MV[t]      = m;
        MV[Cc + t] = rsqrtf(var + 1e-5f);
    }
    __syncthreads();
}

// ---------------- Stage A: conv tower + li/rj precompute ----------------
__global__ __launch_bounds__(256) void conv_tower_kernel(
    const float* __restrict__ t1d, const float* __restrict__ p1w,
    const float* __restrict__ p1b, const float* __restrict__ c1w,
    const float* __restrict__ c2w, const float* __restrict__ pw,
    float* __restrict__ li, float* __restrict__ rj)
{
    extern __shared__ float smem[];
    float* X    = smem;               // 32*512
    float* Y    = X + Cc * Lx;        // 32*512
    float* Z    = Y + Cc * Lx;        // 32*512
    float* WB   = Z + Cc * Lx;        // 3072
    float* RED  = WB + 3072;          // 256
    float* RED2 = RED + 256;          // 256
    float* MV   = RED2 + 256;         // 64

    const int t  = threadIdx.x;
    const int bt = blockIdx.x;

    // input projection: X[c][l] = proj1d_w[c,:] . t1d[bt,l,:] + b[c]
    for (int idx = t; idx < Cc * Lx; idx += 256) {
        int c = idx >> 9, l = idx & (Lx - 1);
        const float* trow = t1d + (size_t)(bt * Lx + l) * F1;
        const float* wrow = p1w + c * F1;
        float acc = p1b[c];
        for (int f = 0; f < F1; ++f) acc = fmaf(wrow[f], trow[f], acc);
        X[idx] = acc;
    }
    __syncthreads();

    for (int blk = 0; blk < 4; ++blk) {
        // y = elu(IN(conv1(x)))
        for (int idx = t; idx < Cc * Cc * 3; idx += 256) WB[idx] = c1w[blk * Cc * Cc * 3 + idx];
        __syncthreads();
        conv3(X, Y, WB, t);
        __syncthreads();
        inorm_stats(Y, RED, RED2, MV, t);
        for (int idx = t; idx < Cc * Lx; idx += 256) {
            int c = idx >> 9;
            Y[idx] = elu1((Y[idx] - MV[c]) * MV[Cc + c]);
        }
        __syncthreads();
        // z = IN(conv2(y)); x = elu(x + z)
        for (int idx = t; idx < Cc * Cc * 3; idx += 256) WB[idx] = c2w[blk * Cc * Cc * 3 + idx];
        __syncthreads();
        conv3(Y, Z, WB, t);
        __syncthreads();
        inorm_stats(Z, RED, RED2, MV, t);
        for (int idx = t; idx < Cc * Lx; idx += 256) {
            int c = idx >> 9;
            float zn = (Z[idx] - MV[c]) * MV[Cc + c];
            X[idx] = elu1(X[idx] + zn);
        }
        __syncthreads();
    }

    // hoisted pair terms: li[bt,i,o] = Wl[o,:].emb[i], rj likewise (Wr)
    for (int idx = t; idx < Lx * Kk; idx += 256) {
        int i = idx >> 6, o = idx & 63;
        const float* wl = pw + o * PWROW + F2;       // proj_w[o, F2 : F2+C]
        const float* wr = wl + Cc;                   // proj_w[o, F2+C : F2+2C]
        float aL = 0.f, aR = 0.f;
        for (int c = 0; c < Cc; ++c) {
            float xv = X[(c << 9) + i];
            aL = fmaf(wl[c], xv, aL);
            aR = fmaf(wr[c], xv, aR);
        }
        size_t g = ((size_t)(bt * Lx) + i) * Kk + o;
        li[g] = aL;
        rj[g] = aR;
    }
}

// ---------------- Stage B: pair projection via WMMA f32 16x16x4 ----------------
// grid: (jg=8, i=512, bt=4), block: 128 threads = 4 waves.
// Wave w computes out[bt, i, jg*64 + w*16 .. +16, 0..64).
__global__ __launch_bounds__(128) void pair_kernel(
    const float* __restrict__ t2d, const float* __restrict__ pw,
    const float* __restrict__ pb,  const float* __restrict__ li,
    const float* __restrict__ rj,  float* __restrict__ out)
{
    __shared__ v2f   Wp[22 * 64];          // B operand: {W2d[n][2fp], W2d[n][2fp+1]}
    __shared__ float Rj[64 * 64];          // right-term tile for this block's 64 j's
    __shared__ float Base[64];             // li[bt,i,o] + bias[o]
    __shared__ float Wsep[64];
    __shared__ float PeI[32];              // pe_half[i][0..31]
    __shared__ __align__(16) float Ot[4][16 * 64];  // per-wave output staging

    const int t  = threadIdx.x;
    const int jg = blockIdx.x;
    const int i  = blockIdx.y;
    const int bt = blockIdx.z;

    for (int idx = t; idx < 22 * 64; idx += 128) {
        int fp = idx >> 6, n = idx & 63;
        const float* r = pw + n * PWROW + 2 * fp;
        v2f v; v.x = r[0]; v.y = r[1];
        Wp[idx] = v;
    }
    for (int idx = t; idx < 64 * 64; idx += 128) {
        int jl = idx >> 6, o = idx & 63;
        Rj[idx] = rj[((size_t)(bt * Lx) + jg * 64 + jl) * Kk + o];
    }
    if (t < 64) {
        Base[t] = li[((size_t)(bt * Lx) + i) * Kk + t] + pb[t];
        Wsep[t] = pw[t * PWROW + (F2 + 2 * Cc)];
    }
    if (t < 32) {
        int c = t & 15;                      // div_term[c] = 10000^(-c/16)
        float ang = (float)i * __expf(-0.5756462732485115f * (float)c);
        PeI[t] = (t < 16) ? __sinf(ang) : __cosf(ang);
    }
    __syncthreads();

    const int wid  = t >> 5;
    const int lane = t & 31;
    const int m    = lane & 15;              // A row / B column within 16
    const int hi   = lane >> 4;              // K half select (ISA f32 A/B layout)
    const int jbase = jg * 64 + wid * 16;
    const float* Arow = t2d + ((size_t)(bt * Lx + i) * Lx + (jbase + m)) * F2;

    v8f acc[4] = {};
#pragma unroll
    for (int ks = 0; ks < 11; ++ks) {        // K = 44 = 11 * 4
        // A 16x4 f32 layout: lane holds row (lane&15), K = 4*ks + 2*hi + {0,1}
        // t2d is read exactly once -> non-temporal (don't pollute L2)
        v2f a = __builtin_nontemporal_load((const v2f*)(Arow + 4 * ks + 2 * hi));
#pragma unroll
        for (int nt = 0; nt < 4; ++nt) {
            // B 4x16 f32 layout: VGPR v, lanes<16 -> K=v, lanes>=16 -> K=v+2
            v2f b = Wp[(2 * ks + hi) * 64 + nt * 16 + m];
            acc[nt] = __builtin_amdgcn_wmma_f32_16x16x4_f32(
                false, a, false, b, (short)0, acc[nt], false, false);
        }
    }

    // Epilogue: C/D layout — VGPR v, lanes<16 -> M=v, lanes>=16 -> M=v+8
#pragma unroll
    for (int nt = 0; nt < 4; ++nt) {
#pragma unroll
        for (int v = 0; v < 8; ++v) {
            int mr = v + 8 * hi;
            int o  = nt * 16 + m;
            int j  = jbase + mr;
            float d   = fabsf((float)(i - j));
            float val = acc[nt][v] + Base[o] + Rj[(wid * 16 + mr) * 64 + o]
                      + __logf(d + 1.0f) * Wsep[o];
            if (bt == 0) {                   // batch_mask: only bt==0 gets PE
                float pe;
                if (o < 32) {
                    pe = PeI[o];
                } else {
                    int oc = o - 32;
                    int c  = oc & 15;
                    float ang = (float)j * __expf(-0.5756462732485115f * (float)c);
                    pe = (oc < 16) ? __sinf(ang) : __cosf(ang);
                }
                val += pe;
            }
            Ot[wid][mr * 64 + o] = val;
        }
    }
    __syncthreads();

    // Coalesced store: each wave's 16x64 tile is a contiguous 4KB in memory.
    // Output is write-once streaming -> non-temporal stores.
    float* gout = out + ((size_t)(bt * Lx + i) * Lx + jbase) * Kk;
#pragma unroll
    for (int q = 0; q < 8; ++q) {
        int c = q * 32 + lane;
        v4f v4 = *(const v4f*)&Ot[wid][c * 4];
        __builtin_nontemporal_store(v4, (v4f*)(gout + c * 4));
    }
}

extern "C" void kernel_launch(void* const* d_in, const int* in_sizes, int n_in,
                              void* d_out, int out_size, void* d_ws, size_t ws_size,
                              hipStream_t stream) {
    const float* t1d = (const float*)d_in[0];
    const float* t2d = (const float*)d_in[1];
    const float* p1w = (const float*)d_in[2];
    const float* p1b = (const float*)d_in[3];
    const float* c1w = (const float*)d_in[4];
    const float* c2w = (const float*)d_in[5];
    const float* pw  = (const float*)d_in[6];
    const float* pb  = (const float*)d_in[7];
    float* out = (float*)d_out;

    float* li = (float*)d_ws;                 // 4*512*64 floats
    float* rj = li + 4 * Lx * Kk;             // 4*512*64 floats (1 MB total)

    // 3 activation buffers + weights + reduction scratch
    size_t smemA = (size_t)(3 * Cc * Lx + 3072 + 256 + 256 + 64) * sizeof(float);

    conv_tower_kernel<<<dim3(4), dim3(256), smemA, stream>>>(
        t1d, p1w, p1b, c1w, c2w, pw, li, rj);

    pair_kernel<<<dim3(8, Lx, 4), dim3(128), 0, stream>>>(
        t2d, pw, pb, li, rj, out);
}